// AttentionRNN_91302414778804
// MI455X (gfx1250) — compile-verified
//
#include <hip/hip_runtime.h>
#include <hip/hip_bf16.h>

// Model dims
#define V_  128
#define E_  32
#define H_  128
#define B_  4
#define S_  512
#define G4H 512          // 4*H
#define HS  132          // padded LDS stride for h (bank-conflict avoidance)

typedef __attribute__((ext_vector_type(2))) float v2f;
typedef __attribute__((ext_vector_type(8))) float v8f;

__device__ __forceinline__ v8f wmma_f32(v2f a, v2f b, v8f c) {
    // V_WMMA_F32_16X16X4_F32 : D(16x16,f32) = A(16x4,f32) * B(4x16,f32) + C
    return __builtin_amdgcn_wmma_f32_16x16x4_f32(
        /*neg_a=*/false, a, /*neg_b=*/false, b,
        /*c_mod=*/(short)0, c, /*reuse_a=*/false, /*reuse_b=*/false);
}

__device__ __forceinline__ float sigmf(float x) {
    return 1.0f / (1.0f + __expf(-x));
}

// ---------------------------------------------------------------------------
// Kernel 1: Gx[b*S+t, n] = (emb[x[b,t]] @ W_ih^T)[n] + b_ih[n] + b_hh[n]
// M = 2048 (128 tiles), N = 512 (32 tiles), K = 32. One wave per 16x16 tile.
// ---------------------------------------------------------------------------
__global__ void gx_kernel(const int* __restrict__ x, const float* __restrict__ emb,
                          const float* __restrict__ Wih, const float* __restrict__ bih,
                          const float* __restrict__ bhh, float* __restrict__ Gx) {
    const int tid    = threadIdx.x;
    const int lane   = tid & 31, wv = tid >> 5;
    const int laneLo = lane & 15, laneHi = lane >> 4;
    const int tile   = blockIdx.x * 8 + wv;     // 0..4095
    const int mtile  = tile >> 5;               // 0..127
    const int ntile  = tile & 31;               // 0..31
    const int m0 = mtile * 16, n0 = ntile * 16;

    const int tok = x[m0 + laneLo];
    const float* arow = emb + (size_t)tok * E_;             // A row (gathered embedding)
    const float* brow = Wih + (size_t)(n0 + laneLo) * E_;   // B[k][n] = W_ih[n][k]

    v8f acc = {};
#pragma unroll
    for (int k0 = 0; k0 < E_; k0 += 4) {
        const int kk = k0 + 2 * laneHi;
        v2f a, b;
        a.x = arow[kk];  a.y = arow[kk + 1];
        b.x = brow[kk];  b.y = brow[kk + 1];
        acc = wmma_f32(a, b, acc);
    }
    const float bias = bih[n0 + laneLo] + bhh[n0 + laneLo];
#pragma unroll
    for (int r = 0; r < 8; ++r) {
        const int m = r + 8 * laneHi;           // C/D layout: VGPR r -> row r (+8 hi half)
        Gx[(size_t)(m0 + m) * G4H + n0 + laneLo] = acc[r] + bias;
    }
}

// ---------------------------------------------------------------------------
// Kernel 2: sequential LSTM scan on ONE workgroup (8 waves).
// h kept as a 16x128 LDS tile (rows 0..3 = batch, rows 4..15 zero) so the
// M=16 WMMA tile is directly valid. Per step: 8 waves x 4 N-tiles x 32 WMMAs.
// ---------------------------------------------------------------------------
__global__ void lstm_kernel(const float* __restrict__ Gx, const float* __restrict__ Whh,
                            float* __restrict__ outh, float* __restrict__ outTail) {
    __shared__ float sH[16 * HS];
    __shared__ float sC[B_ * H_];
    __shared__ float sG[B_ * G4H];

    const int tid = threadIdx.x;
    for (int i = tid; i < 16 * HS; i += 256) sH[i] = 0.0f;
    for (int i = tid; i < B_ * H_; i += 256) sC[i] = 0.0f;
    __syncthreads();

    const int lane   = tid & 31, wv = tid >> 5;
    const int laneLo = lane & 15, laneHi = lane >> 4;

    for (int t = 0; t < S_; ++t) {
        // gates = h @ W_hh^T  (+ precomputed Gx incl. biases)
#pragma unroll
        for (int nt = 0; nt < 4; ++nt) {
            const int n0 = wv * 64 + nt * 16;
            const float* brow = Whh + (size_t)(n0 + laneLo) * H_;  // B[k][n] = W_hh[n][k]
            v8f acc = {};
#pragma unroll 8
            for (int k0 = 0; k0 < H_; k0 += 4) {
                const int kk = k0 + 2 * laneHi;
                v2f a, b;
                a.x = sH[laneLo * HS + kk];  a.y = sH[laneLo * HS + kk + 1];
                b.x = brow[kk];              b.y = brow[kk + 1];
                acc = wmma_f32(a, b, acc);
            }
            if (laneHi == 0) {               // rows 0..3 are the live batch rows
#pragma unroll
                for (int r = 0; r < B_; ++r) {
                    const int n = n0 + laneLo;
                    sG[r * G4H + n] = acc[r] + Gx[((size_t)(r * S_ + t)) * G4H + n];
                }
            }
        }
        __syncthreads();

        // pointwise LSTM cell update: 512 (b,h) pairs over 256 threads
        for (int idx = tid; idx < B_ * H_; idx += 256) {
            const int b = idx >> 7, h = idx & (H_ - 1);
            const float* g = sG + b * G4H;
            const float gi = g[h], gf = g[H_ + h], gg = g[2 * H_ + h], go = g[3 * H_ + h];
            const float c  = sigmf(gf) * sC[b * H_ + h] + sigmf(gi) * tanhf(gg);
            const float hn = sigmf(go) * tanhf(c);
            sC[b * H_ + h] = c;
            sH[b * HS + h] = hn;
            outh[((size_t)(b * S_ + t)) * H_ + h] = hn;
            if (t == S_ - 1) {
                outTail[b * H_ + h]            = hn;   // hT
                outTail[B_ * H_ + b * H_ + h]  = c;    // cT
            }
        }
        __syncthreads();
    }
}

// ---------------------------------------------------------------------------
// Kernel 3: Q = out@Wq^T + bq ; K = out@Wk^T + bk.
// M = 2048 (128 tiles), N = 128 (8 tiles), K = 128, x2 matrices = 2048 tiles.
// ---------------------------------------------------------------------------
__global__ void qk_kernel(const float* __restrict__ outh,
                          const float* __restrict__ Wq, const float* __restrict__ bq,
                          const float* __restrict__ Wk, const float* __restrict__ bk,
                          float* __restrict__ Qb, float* __restrict__ Kb) {
    const int tid    = threadIdx.x;
    const int lane   = tid & 31, wv = tid >> 5;
    const int laneLo = lane & 15, laneHi = lane >> 4;
    const int tile   = blockIdx.x * 8 + wv;     // 0..2047
    const int mat    = tile >> 10;              // 0 = Q, 1 = K
    const int tt     = tile & 1023;
    const int mtile  = tt >> 3, ntile = tt & 7;
    const int m0 = mtile * 16, n0 = ntile * 16;

    const float* W    = mat ? Wk : Wq;
    const float* bias = mat ? bk : bq;
    float*       D    = mat ? Kb : Qb;

    const float* arow = outh + (size_t)(m0 + laneLo) * H_;
    const float* brow = W    + (size_t)(n0 + laneLo) * H_;  // B[k][n] = W[n][k]

    v8f acc = {};
#pragma unroll 8
    for (int k0 = 0; k0 < H_; k0 += 4) {
        const int kk = k0 + 2 * laneHi;
        v2f a, b;
        a.x = arow[kk];  a.y = arow[kk + 1];
        b.x = brow[kk];  b.y = brow[kk + 1];
        acc = wmma_f32(a, b, acc);
    }
    const float bs = bias[n0 + laneLo];
#pragma unroll
    for (int r = 0; r < 8; ++r) {
        const int m = r + 8 * laneHi;
        D[(size_t)(m0 + m) * H_ + n0 + laneLo] = acc[r] + bs;
    }
}

// ---------------------------------------------------------------------------
// Kernel 4: one 128-thread block per (b, s): Bahdanau scores + causal softmax
// + context + final projection into logits[b,s,:].
// ---------------------------------------------------------------------------
__global__ void attn_kernel(const float* __restrict__ outh, const float* __restrict__ Qb,
                            const float* __restrict__ Kb, const float* __restrict__ vvec,
                            const float* __restrict__ Wf, const float* __restrict__ bf,
                            float* __restrict__ logits) {
    __shared__ float sq[H_], sO[H_], sCtx[H_], red[128];
    __shared__ float sS[S_];
    const int tid = threadIdx.x;
    const int bs  = blockIdx.x;
    const int b   = bs >> 9, s = bs & (S_ - 1);

    sq[tid] = Qb[(size_t)bs * H_ + tid];
    sO[tid] = outh[(size_t)bs * H_ + tid];
    __syncthreads();

    // scores for keys t = tid, tid+128, ... (causal: t <= s)
    float lmax = -3.402823466e38f;
    for (int t = tid; t <= s; t += 128) {
        const float* kr = Kb + ((size_t)(b * S_ + t)) * H_;
        float sc = 0.0f;
#pragma unroll 4
        for (int h = 0; h < H_; ++h)
            sc += vvec[h] * tanhf(sq[h] + kr[h]);
        sS[t] = sc;
        lmax  = fmaxf(lmax, sc);
    }
    red[tid] = lmax;
    __syncthreads();
    for (int off = 64; off; off >>= 1) {
        if (tid < off) red[tid] = fmaxf(red[tid], red[tid + off]);
        __syncthreads();
    }
    const float mx = red[0];
    __syncthreads();

    float lsum = 0.0f;
    for (int t = tid; t <= s; t += 128) {
        const float e = __expf(sS[t] - mx);
        sS[t] = e;
        lsum += e;
    }
    red[tid] = lsum;
    __syncthreads();
    for (int off = 64; off; off >>= 1) {
        if (tid < off) red[tid] += red[tid + off];
        __syncthreads();
    }
    const float inv = 1.0f / red[0];
    __syncthreads();

    // context: thread = hidden channel h (coalesced column reads of outh)
    float acc = 0.0f;
    for (int t = 0; t <= s; ++t)
        acc += sS[t] * outh[((size_t)(b * S_ + t)) * H_ + tid];
    sCtx[tid] = acc * inv;
    __syncthreads();

    // logits: thread = vocab row vv;  [out, ctx] @ Wf^T + bf
    const float* wr = Wf + (size_t)tid * (2 * H_);
    float lg = bf[tid];
#pragma unroll 4
    for (int h = 0; h < H_; ++h)
        lg += sO[h] * wr[h] + sCtx[h] * wr[H_ + h];
    logits[(size_t)bs * V_ + tid] = lg;
}

// ---------------------------------------------------------------------------
extern "C" void kernel_launch(void* const* d_in, const int* in_sizes, int n_in,
                              void* d_out, int out_size, void* d_ws, size_t ws_size,
                              hipStream_t stream) {
    (void)in_sizes; (void)n_in; (void)out_size; (void)ws_size;
    const int*   x   = (const int*)  d_in[0];
    const float* emb = (const float*)d_in[1];
    const float* Wih = (const float*)d_in[2];
    const float* Whh = (const float*)d_in[3];
    const float* bih = (const float*)d_in[4];
    const float* bhh = (const float*)d_in[5];
    const float* Wq  = (const float*)d_in[6];
    const float* bq  = (const float*)d_in[7];
    const float* Wk  = (const float*)d_in[8];
    const float* bk  = (const float*)d_in[9];
    const float* v   = (const float*)d_in[10];
    const float* Wf  = (const float*)d_in[11];
    const float* bf  = (const float*)d_in[12];

    float* out = (float*)d_out;                 // logits | hT | cT
    float* ws  = (float*)d_ws;
    float* Gx   = ws;                           // 2048*512  = 1,048,576 f
    float* outh = Gx   + (size_t)2048 * 512;    // 2048*128
    float* Qb   = outh + (size_t)2048 * 128;    // 2048*128
    float* Kb   = Qb   + (size_t)2048 * 128;    // 2048*128  (total ~7 MB)

    gx_kernel  <<<512,  256, 0, stream>>>(x, emb, Wih, bih, bhh, Gx);
    lstm_kernel<<<1,    256, 0, stream>>>(Gx, Whh, outh, out + (size_t)B_ * S_ * V_);
    qk_kernel  <<<256,  256, 0, stream>>>(outh, Wq, bq, Wk, bk, Qb, Kb);
    attn_kernel<<<2048, 128, 0, stream>>>(outh, Qb, Kb, v, Wf, bf, out);
}